// SimpleNN_13322988552355
// MI455X (gfx1250) — compile-verified
//
#include <hip/hip_runtime.h>
#include <hip/hip_bf16.h>

typedef __attribute__((ext_vector_type(2))) float v2f;
typedef __attribute__((ext_vector_type(4))) float v4f;
typedef __attribute__((ext_vector_type(8))) float v8f;

#define B_ 4096
#define S_ 4096
#define H_ 3
#define V_ 2
#define L_ 16                 // chunk length in timesteps
#define NC (S_ / L_)          // 256 chunks per batch row
#define ROWSTRIDE (S_ * V_)   // 8192 floats per batch row

// workspace layout (in floats)
#define OFF_F   0       // 32x32 fused chunk operator
#define OFF_E   1024    // 32x16 (cols 0..2 valid) local-end operator
#define OFF_QB  1536    // 4x32  carry->output operator (row 3 = 0)
#define OFF_A16 1664    // 3x3 A^16
#define OFF_HIN 2048    // B * NC * 4 floats: chunk summaries, then carries in place

// ---------------------------------------------------------------------------
// Pass 0: build the constant operator tables from W_h, W_i, W_o (trivial work)
// ---------------------------------------------------------------------------
__global__ void precompute_ops(const float* __restrict__ Wh,
                               const float* __restrict__ Wi,
                               const float* __restrict__ Wo,
                               float* __restrict__ ws) {
    if (threadIdx.x != 0 || blockIdx.x != 0) return;
    float Ap[L_ + 1][3][3];
    for (int r = 0; r < 3; r++)
        for (int c = 0; c < 3; c++) Ap[0][r][c] = (r == c) ? 1.f : 0.f;
    for (int k = 1; k <= L_; k++)
        for (int r = 0; r < 3; r++)
            for (int c = 0; c < 3; c++) {
                float s = 0.f;
                for (int m = 0; m < 3; m++) s += Ap[k - 1][r][m] * Wh[m * 3 + c];
                Ap[k][r][c] = s;
            }
    // F[(j*2+w)][(i*2+v)] = sum_{a,d} Wi[a][w] * A^{i-j}[a][d] * Wo[v][d]  (j<=i)
    float* F = ws + OFF_F;
    for (int j = 0; j < L_; j++)
        for (int w = 0; w < V_; w++)
            for (int i = 0; i < L_; i++)
                for (int v = 0; v < V_; v++) {
                    float s = 0.f;
                    if (j <= i)
                        for (int a = 0; a < 3; a++)
                            for (int d = 0; d < 3; d++)
                                s += Wi[a * 2 + w] * Ap[i - j][a][d] * Wo[v * 3 + d];
                    F[(j * 2 + w) * 32 + (i * 2 + v)] = s;
                }
    // E[(j*2+w)][d] = sum_a Wi[a][w] * A^{15-j}[a][d]   (padded to 16 cols)
    float* E = ws + OFF_E;
    for (int j = 0; j < L_; j++)
        for (int w = 0; w < V_; w++)
            for (int d = 0; d < 16; d++) {
                float s = 0.f;
                if (d < 3)
                    for (int a = 0; a < 3; a++) s += Wi[a * 2 + w] * Ap[L_ - 1 - j][a][d];
                E[(j * 2 + w) * 16 + d] = s;
            }
    // QB[d0][(i*2+v)] = sum_d A^{i+1}[d0][d] * Wo[v][d]   (row 3 = 0)
    float* QB = ws + OFF_QB;
    for (int d0 = 0; d0 < 4; d0++)
        for (int i = 0; i < L_; i++)
            for (int v = 0; v < V_; v++) {
                float s = 0.f;
                if (d0 < 3)
                    for (int d = 0; d < 3; d++) s += Ap[i + 1][d0][d] * Wo[v * 3 + d];
                QB[d0 * 32 + (i * 2 + v)] = s;
            }
    float* A16 = ws + OFF_A16;
    for (int r = 0; r < 3; r++)
        for (int c = 0; c < 3; c++) A16[r * 3 + c] = Ap[L_][r][c];
}

// ---------------------------------------------------------------------------
// Pass 1: per (16-row batch tile, chunk) local end-state  hloc = X(16x32)*E
// one wave per (btile, chunk-group of 16); 8 f32 WMMAs per chunk
// ---------------------------------------------------------------------------
__global__ void pass1_local_end(const float* __restrict__ x,
                                const float* __restrict__ ws_c,
                                float* __restrict__ hin) {
    const int lane  = threadIdx.x & 31;
    const int wid   = blockIdx.x * (blockDim.x >> 5) + (threadIdx.x >> 5);
    const int btile = wid >> 4;
    const int cgrp  = wid & 15;
    const int half  = lane >> 4;   // 0: K lo, 1: K hi (A & B operand halves)
    const int lrow  = lane & 15;   // A: row M, B: col N

    // Preload B operands (E slices), resident in VGPRs across the chunk loop.
    const float* E = ws_c + OFF_E;
    v2f be[8];
#pragma unroll
    for (int s = 0; s < 8; s++) {
        int k0 = s * 4 + 2 * half;
        be[s][0] = E[k0 * 16 + lrow];
        be[s][1] = E[(k0 + 1) * 16 + lrow];
    }

    const float* xrow = x + (size_t)(btile * 16 + lrow) * ROWSTRIDE;

    for (int cc = 0; cc < 16; cc++) {
        int c = cgrp * 16 + cc;
        const float* xc = xrow + c * 32 + 2 * half;
        v8f acc = {};
#pragma unroll
        for (int s = 0; s < 8; s++) {
            v2f a = *(const v2f*)(xc + s * 4);   // A-slice: X[:, 4s..4s+3]
            acc = __builtin_amdgcn_wmma_f32_16x16x4_f32(
                false, a, false, be[s], (short)0, acc, false, false);
        }
        // D cols 0..3 hold [h0,h1,h2,0]; store as float4 per (b,c)
        if (lrow < 4) {
#pragma unroll
            for (int j = 0; j < 8; j++) {
                int row = j + 8 * half;
                hin[((size_t)(btile * 16 + row) * NC + c) * 4 + lrow] = acc[j];
            }
        }
    }
}

// ---------------------------------------------------------------------------
// Pass 2: per-row sequential carry scan over chunks (in place), emit h_final
// ---------------------------------------------------------------------------
__global__ void pass2_scan(float* __restrict__ hin,
                           const float* __restrict__ ws_c,
                           float* __restrict__ hfinal) {
    int b = blockIdx.x * blockDim.x + threadIdx.x;
    if (b >= B_) return;
    const float* A16 = ws_c + OFF_A16;
    float a[9];
#pragma unroll
    for (int i = 0; i < 9; i++) a[i] = A16[i];
    float c0 = 0.f, c1 = 0.f, c2 = 0.f;
    float* p = hin + (size_t)b * NC * 4;
    for (int c = 0; c < NC; c++) {
        v4f t = *(v4f*)(p + c * 4);
        v4f w; w[0] = c0; w[1] = c1; w[2] = c2; w[3] = 0.f;
        *(v4f*)(p + c * 4) = w;   // slot now holds h_in for chunk c
        float n0 = c0 * a[0] + c1 * a[3] + c2 * a[6] + t[0];
        float n1 = c0 * a[1] + c1 * a[4] + c2 * a[7] + t[1];
        float n2 = c0 * a[2] + c1 * a[5] + c2 * a[8] + t[2];
        c0 = n0; c1 = n1; c2 = n2;
    }
    hfinal[b * 3 + 0] = c0; hfinal[b * 3 + 1] = c1; hfinal[b * 3 + 2] = c2;
}

// ---------------------------------------------------------------------------
// Pass 3: O(16x32) = X(16x32)*F(32x32) + Hin(16x4)*QB(4x32); 18 WMMAs/chunk
// ---------------------------------------------------------------------------
__global__ void pass3_output(const float* __restrict__ x,
                             const float* __restrict__ ws_c,
                             const float* __restrict__ hin,
                             float* __restrict__ out) {
    const int lane  = threadIdx.x & 31;
    const int wid   = blockIdx.x * (blockDim.x >> 5) + (threadIdx.x >> 5);
    const int btile = wid >> 4;
    const int cgrp  = wid & 15;
    const int half  = lane >> 4;
    const int lrow  = lane & 15;

    const float* F  = ws_c + OFF_F;
    const float* QB = ws_c + OFF_QB;
    v2f bf0[8], bf1[8], q0, q1;
#pragma unroll
    for (int s = 0; s < 8; s++) {
        int k0 = s * 4 + 2 * half;
        bf0[s][0] = F[k0 * 32 + lrow];        bf0[s][1] = F[(k0 + 1) * 32 + lrow];
        bf1[s][0] = F[k0 * 32 + 16 + lrow];   bf1[s][1] = F[(k0 + 1) * 32 + 16 + lrow];
    }
    {
        int k0 = 2 * half;
        q0[0] = QB[k0 * 32 + lrow];        q0[1] = QB[(k0 + 1) * 32 + lrow];
        q1[0] = QB[k0 * 32 + 16 + lrow];   q1[1] = QB[(k0 + 1) * 32 + 16 + lrow];
    }

    const float* xrow = x + (size_t)(btile * 16 + lrow) * ROWSTRIDE;
    const float* hrow = hin + (size_t)(btile * 16 + lrow) * NC * 4;

    for (int cc = 0; cc < 16; cc++) {
        int c = cgrp * 16 + cc;
        const float* xc = xrow + c * 32 + 2 * half;
        v8f d0 = {}, d1 = {};
        // carry contribution: Hin(16x4, col3=0) x QB(4x16) per N-tile
        v2f hc = *(const v2f*)(hrow + c * 4 + 2 * half);
        d0 = __builtin_amdgcn_wmma_f32_16x16x4_f32(false, hc, false, q0, (short)0, d0, false, false);
        d1 = __builtin_amdgcn_wmma_f32_16x16x4_f32(false, hc, false, q1, (short)0, d1, false, false);
#pragma unroll
        for (int s = 0; s < 8; s++) {       // two independent accumulator chains
            v2f a = *(const v2f*)(xc + s * 4);
            d0 = __builtin_amdgcn_wmma_f32_16x16x4_f32(false, a, false, bf0[s], (short)0, d0, false, false);
            d1 = __builtin_amdgcn_wmma_f32_16x16x4_f32(false, a, false, bf1[s], (short)0, d1, false, false);
        }
        if (cc < 15) __builtin_prefetch(xc + 32, 0, 1);  // next chunk -> global_prefetch_b8
        // D: VGPR j <-> rows j / j+8, lane%16 <-> col; 64B coalesced segments
#pragma unroll
        for (int j = 0; j < 8; j++) {
            size_t base = (size_t)(btile * 16 + j + 8 * half) * ROWSTRIDE + (size_t)c * 32;
            out[base + lrow]      = d0[j];
            out[base + 16 + lrow] = d1[j];
        }
    }
}

// ---------------------------------------------------------------------------
// Fallback if workspace is too small: plain per-row sequential scan
// ---------------------------------------------------------------------------
__global__ void fallback_seq(const float* __restrict__ x,
                             const float* __restrict__ Wh,
                             const float* __restrict__ Wi,
                             const float* __restrict__ Wo,
                             float* __restrict__ out,
                             float* __restrict__ hfinal) {
    int b = blockIdx.x * blockDim.x + threadIdx.x;
    if (b >= B_) return;
    float wh[9], wi[6], wo[6];
#pragma unroll
    for (int i = 0; i < 9; i++) wh[i] = Wh[i];
#pragma unroll
    for (int i = 0; i < 6; i++) wi[i] = Wi[i];
#pragma unroll
    for (int i = 0; i < 6; i++) wo[i] = Wo[i];
    const float* xr = x + (size_t)b * ROWSTRIDE;
    float* orow = out + (size_t)b * ROWSTRIDE;
    float h0 = 0.f, h1 = 0.f, h2 = 0.f;
    for (int t = 0; t < S_; t++) {
        float x0 = xr[t * 2], x1 = xr[t * 2 + 1];
        float n0 = h0 * wh[0] + h1 * wh[3] + h2 * wh[6] + x0 * wi[0] + x1 * wi[1];
        float n1 = h0 * wh[1] + h1 * wh[4] + h2 * wh[7] + x0 * wi[2] + x1 * wi[3];
        float n2 = h0 * wh[2] + h1 * wh[5] + h2 * wh[8] + x0 * wi[4] + x1 * wi[5];
        h0 = n0; h1 = n1; h2 = n2;
        orow[t * 2]     = h0 * wo[0] + h1 * wo[1] + h2 * wo[2];
        orow[t * 2 + 1] = h0 * wo[3] + h1 * wo[4] + h2 * wo[5];
    }
    hfinal[b * 3 + 0] = h0; hfinal[b * 3 + 1] = h1; hfinal[b * 3 + 2] = h2;
}

extern "C" void kernel_launch(void* const* d_in, const int* in_sizes, int n_in,
                              void* d_out, int out_size, void* d_ws, size_t ws_size,
                              hipStream_t stream) {
    (void)in_sizes; (void)n_in; (void)out_size;
    const float* x  = (const float*)d_in[0];
    const float* Wh = (const float*)d_in[1];
    const float* Wi = (const float*)d_in[2];
    const float* Wo = (const float*)d_in[3];
    float* out    = (float*)d_out;
    float* hfinal = out + (size_t)B_ * S_ * V_;   // h_final concatenated after output

    const size_t need = ((size_t)OFF_HIN + (size_t)B_ * NC * 4) * sizeof(float);
    if (ws_size < need) {
        fallback_seq<<<(B_ + 255) / 256, 256, 0, stream>>>(x, Wh, Wi, Wo, out, hfinal);
        return;
    }

    float* ws  = (float*)d_ws;
    float* hin = ws + OFF_HIN;

    precompute_ops<<<1, 32, 0, stream>>>(Wh, Wi, Wo, ws);

    // (B/16) batch tiles x 16 chunk-groups = 4096 waves; 8 waves per block
    const int waves  = (B_ / 16) * 16;
    const int blocks = waves * 32 / 256;
    pass1_local_end<<<blocks, 256, 0, stream>>>(x, ws, hin);
    pass2_scan<<<B_ / 256, 256, 0, stream>>>(hin, ws, hfinal);
    pass3_output<<<blocks, 256, 0, stream>>>(x, ws, hin, out);
}